// CFConv_83373905150295
// MI455X (gfx1250) — compile-verified
//
#include <hip/hip_runtime.h>

typedef __attribute__((ext_vector_type(2))) float v2f;
typedef __attribute__((ext_vector_type(4))) float v4f;
typedef __attribute__((ext_vector_type(8))) float v8f;

#define FM 128  // F_IN == N_FM == F_OUT == 128

// ---------------------------------------------------------------------------
// Out[M x 128] = A[M x 128] @ W[128 x 128] (+ bias), f32 WMMA 16x16x4.
// Block: 256 threads = 8 waves. Each block owns 16 rows; each wave owns a
// 16-wide slice of the 128 output columns. A-tile staged through LDS so the
// kernel is safe to run in place (A == Out).
// ---------------------------------------------------------------------------
__global__ __launch_bounds__(256) void gemm128_wmma(
    const float* __restrict__ A,     // M x 128
    const float* __restrict__ W,     // 128 x 128 (row major)
    const float* __restrict__ bias,  // 128 or nullptr
    float* __restrict__ Out,         // M x 128
    int M) {
  __shared__ float Xs[16][FM];

  const int tid = threadIdx.x;
  const int m0  = blockIdx.x * 16;

  // Cooperative stage of the 16x128 activation tile (zero-fill OOB rows).
  for (int idx = tid; idx < 16 * FM; idx += 256) {
    const int r = idx >> 7;
    const int c = idx & (FM - 1);
    const int row = m0 + r;
    Xs[r][c] = (row < M) ? A[(size_t)row * FM + c] : 0.0f;
  }
  __syncthreads();

  const int wave = tid >> 5;          // 0..7 -> output column tile
  const int lane = tid & 31;
  const int n0   = wave * 16;
  const int ln   = lane & 15;         // M (for A) / N (for B,C,D) within tile
  const int kb   = (lane >> 4) << 1;  // lanes 0-15: K base 0; lanes 16-31: K base 2

  // C/D accumulator init (bias broadcast: every VGPR row holds column n0+ln).
  v8f acc;
  const float bv = (bias != nullptr) ? bias[n0 + ln] : 0.0f;
#pragma unroll
  for (int r = 0; r < 8; ++r) acc[r] = bv;

  // K = 128 in steps of 4: 32 chained v_wmma_f32_16x16x4_f32.
#pragma unroll 4
  for (int k0 = 0; k0 < FM; k0 += 4) {
    v2f a, b;
    // A 16x4 f32 fragment: lane ln holds (K=kb, K=kb+1) of row ln.
    a.x = Xs[ln][k0 + kb];
    a.y = Xs[ln][k0 + kb + 1];
    // B 4x16 f32 fragment: lane ln holds rows (kb, kb+1), column n0+ln.
    b.x = W[(size_t)(k0 + kb) * FM + n0 + ln];
    b.y = W[(size_t)(k0 + kb + 1) * FM + n0 + ln];
    acc = __builtin_amdgcn_wmma_f32_16x16x4_f32(
        /*neg_a=*/false, a, /*neg_b=*/false, b,
        /*c_mod=*/(short)0, acc, /*reuse_a=*/false, /*reuse_b=*/false);
  }

  // D layout: VGPR r -> row (r + 8*(lane>=16)), col = n0 + ln.
  // Uniform in-range check: whole 16-row tile either fits or not, so the
  // common path is 8 unguarded, clause-able stores (no per-lane exec churn).
  const int rbase = (lane >> 4) << 3;
  if (m0 + 16 <= M) {
#pragma unroll
    for (int r = 0; r < 8; ++r) {
      Out[(size_t)(m0 + rbase + r) * FM + n0 + ln] = acc[r];
    }
  } else {
#pragma unroll
    for (int r = 0; r < 8; ++r) {
      const int row = m0 + rbase + r;
      if (row < M) Out[(size_t)row * FM + n0 + ln] = acc[r];
    }
  }
}

// ---------------------------------------------------------------------------
// conv[i,:] = sum_{e : seg_i[e]==i} w_ij[e,:] * f[idx_j[e],:]
// One WAVE per atom i (4 atoms per 128-thread block). seg_i is sorted ->
// binary-search the contiguous edge range (wave-uniform). Each lane owns 4
// feature channels: per edge one 512 B/wave non-temporal global_load_b128 of
// the stream-once w_ij row + one b128 read of the L2-resident f row.
// No atomics -> deterministic.
// ---------------------------------------------------------------------------
__global__ __launch_bounds__(128) void cfconv_edges(
    const float* __restrict__ w_ij,   // n_edges x 128
    const int* __restrict__ seg_i,    // n_edges, sorted
    const int* __restrict__ idx_j,    // n_edges
    const float* __restrict__ f,      // n_atoms x 128 (L2 resident)
    float* __restrict__ conv,         // n_atoms x 128
    int n_edges, int n_atoms) {
  const int wave = threadIdx.x >> 5;
  const int lane = threadIdx.x & 31;
  const int i    = blockIdx.x * 4 + wave;  // atom handled by this wave
  if (i >= n_atoms) return;

  // lower_bound(seg_i, i) and lower_bound(seg_i, i+1); uniform across wave.
  int lo = 0, hi = n_edges;
  while (lo < hi) {
    const int mid = (lo + hi) >> 1;
    if (seg_i[mid] < i) lo = mid + 1; else hi = mid;
  }
  const int start = lo;
  hi = n_edges;
  while (lo < hi) {
    const int mid = (lo + hi) >> 1;
    if (seg_i[mid] <= i) lo = mid + 1; else hi = mid;
  }
  const int end = lo;

  v4f acc = {0.0f, 0.0f, 0.0f, 0.0f};
  for (int e = start; e < end; ++e) {
    // Speculative prefetch a few edge-rows ahead (global_prefetch_b8).
    __builtin_prefetch(w_ij + (size_t)(e + 4) * FM + lane * 4, 0, 1);
    const int j  = idx_j[e];  // wave-uniform -> scalar load
    const v4f w  = __builtin_nontemporal_load(
        (const v4f*)(w_ij + (size_t)e * FM) + lane);
    const v4f fj = *((const v4f*)(f + (size_t)j * FM) + lane);
    acc += w * fj;
  }
  *((v4f*)(conv + (size_t)i * FM) + lane) = acc;
}

// ---------------------------------------------------------------------------
extern "C" void kernel_launch(void* const* d_in, const int* in_sizes, int n_in,
                              void* d_out, int out_size, void* d_ws, size_t ws_size,
                              hipStream_t stream) {
  // setup_inputs order: x, w_ij, seg_i, idx_j, seg_i_sum, W_in, W_out, b_out
  const float* x     = (const float*)d_in[0];
  const float* w_ij  = (const float*)d_in[1];
  const int*   seg_i = (const int*)d_in[2];
  const int*   idx_j = (const int*)d_in[3];
  const float* W_in  = (const float*)d_in[5];
  const float* W_out = (const float*)d_in[6];
  const float* b_out = (const float*)d_in[7];
  float*       out   = (float*)d_out;

  const int n_atoms = in_sizes[0] / FM;
  const int n_edges = in_sizes[2];

  float* f    = (float*)d_ws;  // n_atoms*128 f32 (10.2 MB) scratch
  float* conv = out;           // stage conv in d_out; final GEMM runs in place

  const int mblocks = (n_atoms + 15) / 16;
  const int eblocks = (n_atoms + 3) / 4;

  // f = x @ W_in                        (WMMA f32 16x16x4)
  gemm128_wmma<<<mblocks, 256, 0, stream>>>(x, W_in, nullptr, f, n_atoms);
  // conv = segment_sum(w_ij * f[idx_j]) (HBM-stream bound; NT b128 loads)
  cfconv_edges<<<eblocks, 128, 0, stream>>>(w_ij, seg_i, idx_j, f, conv,
                                            n_edges, n_atoms);
  // out = conv @ W_out + b_out          (WMMA, in place on d_out)
  gemm128_wmma<<<mblocks, 256, 0, stream>>>(conv, W_out, b_out, out, n_atoms);
}